// NewSparseHW_86337432584597
// MI455X (gfx1250) — compile-verified
//
#include <hip/hip_runtime.h>
#include <math.h>

typedef float    v2f __attribute__((ext_vector_type(2)));
typedef float    v8f __attribute__((ext_vector_type(8)));
typedef unsigned v4u __attribute__((ext_vector_type(4)));
typedef int      v4i __attribute__((ext_vector_type(4)));
typedef int      v8i __attribute__((ext_vector_type(8)));

#define NBATCH 32
#define NCH    256
#define HH     64
#define WW     64
#define HW     4096
#define KSP    409   // int(0.1 * 4096)
#define KCH    76    // int(0.3 * 256)

__device__ __forceinline__ float block_reduce_sum(float v, float* red, int t) {
  __syncthreads();                 // protect previous user of `red`
  red[t] = v;
  __syncthreads();
  for (int off = 128; off > 0; off >>= 1) {
    if (t < off) red[t] += red[t + off];
    __syncthreads();
  }
  return red[0];
}

// ---------------------------------------------------------------------------
// K1: one block per (n,c) channel. TDM-stage 16KB tile to LDS, exact radix
// top-k on |x| (jax-stable ties), write unmasked sparse_x, emit per-channel
// stats: sparse mass, total |x| mass, row/col entropies.
// ---------------------------------------------------------------------------
__global__ __launch_bounds__(256) void k1_spatial_topk(
    const float* __restrict__ x, float* __restrict__ out,
    float* __restrict__ sparse_ch, float* __restrict__ Sarr,
    float* __restrict__ entArr)
{
  const int nc = blockIdx.x;
  const int t  = threadIdx.x;
  __shared__ float    chan[HW];
  __shared__ unsigned bucket[16];
  __shared__ float    red[256];
  __shared__ unsigned scan[256];
  __shared__ float    rowsum[HH];
  __shared__ float    colsum[WW];
  __shared__ unsigned bc_pref, bc_k;

  const float* src = x + (size_t)nc * HW;

  // ---- Tensor Data Mover: stage 16KB channel tile into LDS (wave 0 only) ----
  if (t < 32) {
    unsigned lds_addr = (unsigned)(unsigned long long)(const void*)chan; // generic LDS ptr low 32 = LDS offset
    unsigned long long ga = (unsigned long long)src;
    v4u g0; v8i g1; v4i g2 = {0, 0, 0, 0}, g3 = {0, 0, 0, 0};
    g0[0] = 1u;                                              // count=1 valid D#
    g0[1] = lds_addr;                                        // lds_addr
    g0[2] = (unsigned)(ga & 0xffffffffull);                  // global_addr[31:0]
    g0[3] = (unsigned)((ga >> 32) & 0x01ffffffull) | 0x80000000u; // addr[56:32] | type=2
    g1[0] = (int)(2u << 16);                                 // data_size = 4B
    g1[1] = (int)(((unsigned)HW & 0xffffu) << 16);           // tensor_dim0 lo16
    g1[2] = (int)(((unsigned)HW >> 16) | (1u << 16));        // tensor_dim0 hi16, tensor_dim1=1
    g1[3] = (int)((unsigned)HW << 16);                       // tile_dim0 = 4096
    g1[4] = 1;                                               // tile_dim1 = 1 (explicit), tile_dim2=0
    g1[5] = (int)HW;                                         // tensor_dim0_stride lo32
    g1[6] = 0;
    g1[7] = 0;
#if __clang_major__ >= 23
    v8i g4 = {0, 0, 0, 0, 0, 0, 0, 0};
    __builtin_amdgcn_tensor_load_to_lds(g0, g1, g2, g3, g4, 0);
#else
    __builtin_amdgcn_tensor_load_to_lds(g0, g1, g2, g3, 0);
#endif
    __builtin_amdgcn_s_wait_tensorcnt(0);
  }
  __syncthreads();

  float    v[16];
  unsigned ab[16];
  const int base = t * 16;           // thread owns 16 contiguous elements (one row each)
#pragma unroll
  for (int j = 0; j < 16; ++j) {
    v[j]  = chan[base + j];
    ab[j] = __float_as_uint(v[j]) & 0x7fffffffu;
  }

  // ---- exact k-th largest |x| via nibble radix-select ----
  unsigned pref = 0, maskhi = 0, kk = KSP;
  for (int shift = 28; shift >= 0; shift -= 4) {
    if (t < 16) bucket[t] = 0u;
    __syncthreads();
#pragma unroll
    for (int j = 0; j < 16; ++j)
      if ((ab[j] & maskhi) == pref) atomicAdd(&bucket[(ab[j] >> shift) & 15u], 1u);
    __syncthreads();
    if (t == 0) {
      unsigned c = 0; int nib = 15;
      for (; nib > 0; --nib) { if (c + bucket[nib] >= kk) break; c += bucket[nib]; }
      bc_pref = pref | ((unsigned)nib << shift);
      bc_k    = kk - c;
    }
    __syncthreads();
    pref = bc_pref; kk = bc_k; maskhi |= (0xFu << shift);
  }
  // pref = exact k-th value bits; kk = #ties to take (lowest indices first)

  unsigned eqc = 0;
#pragma unroll
  for (int j = 0; j < 16; ++j) eqc += (ab[j] == pref) ? 1u : 0u;
  scan[t] = eqc;
  __syncthreads();
  for (int off = 1; off < 256; off <<= 1) {          // Hillis-Steele inclusive scan
    unsigned add = (t >= off) ? scan[t - off] : 0u;
    __syncthreads();
    scan[t] += add;
    __syncthreads();
  }
  unsigned rank = scan[t] - eqc;                     // exclusive prefix of tie count

  float* dst = out + (size_t)nc * HW;
  float selabs = 0.f, rowlocal = 0.f;
  float o[16];
#pragma unroll
  for (int j = 0; j < 16; ++j) {
    bool s;
    if (ab[j] > pref)        s = true;
    else if (ab[j] == pref)  { s = (rank < kk); rank++; }
    else                     s = false;
    float a = __uint_as_float(ab[j]);
    o[j] = s ? v[j] : 0.f;
    selabs   += s ? a : 0.f;
    rowlocal += a;
  }
  float4* d4 = (float4*)dst;
#pragma unroll
  for (int q = 0; q < 4; ++q)
    d4[t * 4 + q] = make_float4(o[4 * q], o[4 * q + 1], o[4 * q + 2], o[4 * q + 3]);

  // ---- row / col |x| sums for locality entropy ----
  if (t < HH) { rowsum[t] = 0.f; colsum[t] = 0.f; }
  __syncthreads();
  atomicAdd(&rowsum[t >> 2], rowlocal);              // thread's 16 elems share one row
  {
    const int c = t & 63, hg = t >> 6;               // 4 threads per column
    float s = 0.f;
#pragma unroll
    for (int i = 0; i < 16; ++i) s += fabsf(chan[(hg * 16 + i) * 64 + c]);
    atomicAdd(&colsum[c], s);
  }

  float sc = block_reduce_sum(selabs, red, t);
  if (t == 0) sparse_ch[nc] = sc;

  float S = block_reduce_sum((t < HH) ? rowsum[t] : 0.f, red, t);
  float rterm = 0.f, cterm = 0.f;
  if (t < HH) {
    float r = rowsum[t]; if (r > 0.f) rterm = r * logf(r);
    float c = colsum[t]; if (c > 0.f) cterm = c * logf(c);
  }
  float sumr = block_reduce_sum(rterm, red, t);
  float sumc = block_reduce_sum(cterm, red, t);
  if (t == 0) {
    float entx = logf(S) - sumr / S;                 // -(sum p log p), p = r/S
    float enty = logf(S) - sumc / S;
    Sarr[nc]   = S;
    entArr[nc] = entx + enty;
  }
}

// ---------------------------------------------------------------------------
// K2: one block per n. Top-KCH channels by sparse mass (ch_prob is a monotone
// transform), jax-stable ties; emit mask, coverage weights, entropy partials.
// ---------------------------------------------------------------------------
__global__ __launch_bounds__(256) void k2_channel_topk(
    const float* __restrict__ sparse_ch, const float* __restrict__ Sarr,
    const float* __restrict__ entArr, float* __restrict__ chmask,
    float* __restrict__ wch, float* __restrict__ entW, float* __restrict__ Ssum)
{
  const int n = blockIdx.x, t = threadIdx.x;
  __shared__ unsigned bucket[16];
  __shared__ float    red[256];
  __shared__ unsigned scan[256];
  __shared__ unsigned bc_pref, bc_k;

  const int idx = n * NCH + t;
  const unsigned ab = __float_as_uint(sparse_ch[idx]) & 0x7fffffffu;

  unsigned pref = 0, maskhi = 0, kk = KCH;
  for (int shift = 28; shift >= 0; shift -= 4) {
    if (t < 16) bucket[t] = 0u;
    __syncthreads();
    if ((ab & maskhi) == pref) atomicAdd(&bucket[(ab >> shift) & 15u], 1u);
    __syncthreads();
    if (t == 0) {
      unsigned c = 0; int nib = 15;
      for (; nib > 0; --nib) { if (c + bucket[nib] >= kk) break; c += bucket[nib]; }
      bc_pref = pref | ((unsigned)nib << shift);
      bc_k    = kk - c;
    }
    __syncthreads();
    pref = bc_pref; kk = bc_k; maskhi |= (0xFu << shift);
  }

  const unsigned eq = (ab == pref) ? 1u : 0u;
  scan[t] = eq;
  __syncthreads();
  for (int off = 1; off < 256; off <<= 1) {
    unsigned add = (t >= off) ? scan[t - off] : 0u;
    __syncthreads();
    scan[t] += add;
    __syncthreads();
  }
  const unsigned excl = scan[t] - eq;
  const bool sel = (ab > pref) || (eq && excl < kk);

  const float S = Sarr[idx];
  chmask[idx] = sel ? 1.f : 0.f;
  wch[idx]    = sel ? (1.f / S) : 0.f;               // coverage weight mask/S

  float Stot = block_reduce_sum(S, red, t);
  float ew   = block_reduce_sum(entArr[idx] * S / Stot, red, t);
  if (t == 0) { entW[n] = ew; Ssum[n] = Stot; }
}

// ---------------------------------------------------------------------------
// K3: coverage via chained V_WMMA_F32_16X16X4_F32.
// cov[p] = sum_c w[c] * |x[c,p]|. Per step: A(16x4)=broadcast w[c0..c0+3],
// B(4x16)=|x| of 4 channels x 16 pixels, C chained. Active 4-channel chunks
// are iterated via two register bitmasks (ctz walk) -> no LDS probes, no
// wasted loads; two independent WMMA chains hide the C RAW latency.
// ---------------------------------------------------------------------------
__global__ __launch_bounds__(256) void k3_coverage_wmma(
    const float* __restrict__ x, const float* __restrict__ wch,
    float* __restrict__ covpart)
{
  const int n = blockIdx.x;
  const int t = threadIdx.x;
  __shared__ float wsh[NCH];
  __shared__ float sk[NCH / 4];
  __shared__ float red[256];

  wsh[t] = wch[n * NCH + t];
  __syncthreads();
  if (t < NCH / 4) sk[t] = wsh[4 * t] + wsh[4 * t + 1] + wsh[4 * t + 2] + wsh[4 * t + 3];
  __syncthreads();

  // Build block-uniform active-chunk bitmasks once (64 chunks of 4 channels).
  unsigned m0 = 0u, m1 = 0u;
#pragma unroll
  for (int i = 0; i < 32; ++i) {
    m0 |= (sk[i]      != 0.f ? 1u : 0u) << i;
    m1 |= (sk[i + 32] != 0.f ? 1u : 0u) << i;
  }

  const int wave = t >> 5;
  const int lane = t & 31;
  const int half = lane >> 4;        // A/B K-split: lanes 0-15 -> K=0,1; 16-31 -> K=2,3
  const int li   = lane & 15;        // output column (pixel in tile)
  const float* xb = x + (size_t)n * NCH * HW;

  float acc = 0.f;
  for (int tile = wave; tile < HW / 16; tile += 8) {
    const int p = tile * 16 + li;
    v8f c0acc = {0.f, 0.f, 0.f, 0.f, 0.f, 0.f, 0.f, 0.f};
    v8f c1acc = {0.f, 0.f, 0.f, 0.f, 0.f, 0.f, 0.f, 0.f};

    for (unsigned m = m0; m != 0u; m &= m - 1u) {    // chunks 0..31
      const int c0 = (int)__builtin_ctz(m) * 4;
      const float* bp = xb + (size_t)(c0 + half * 2) * HW + p;
      v2f a = { wsh[c0 + half * 2], wsh[c0 + half * 2 + 1] };
      v2f b = { fabsf(bp[0]), fabsf(bp[HW]) };
      c0acc = __builtin_amdgcn_wmma_f32_16x16x4_f32(false, a, false, b, (short)0,
                                                    c0acc, false, false);
    }
    for (unsigned m = m1; m != 0u; m &= m - 1u) {    // chunks 32..63
      const int c0 = 128 + (int)__builtin_ctz(m) * 4;
      const float* bp = xb + (size_t)(c0 + half * 2) * HW + p;
      v2f a = { wsh[c0 + half * 2], wsh[c0 + half * 2 + 1] };
      v2f b = { fabsf(bp[0]), fabsf(bp[HW]) };
      c1acc = __builtin_amdgcn_wmma_f32_16x16x4_f32(false, a, false, b, (short)0,
                                                    c1acc, false, false);
    }
    float d = (c0acc[0] + c1acc[0]) - 1.0f;          // all D rows equal; row 0 lanes<16
    acc += (half == 0) ? d * d : 0.0f;               // count each pixel once
  }
  float tot = block_reduce_sum(acc, red, t);
  if (t == 0) covpart[n] = tot;
}

// ---------------------------------------------------------------------------
// K4: zero sparse_x for channels failing the channel mask (STE forward == mask).
// ---------------------------------------------------------------------------
__global__ __launch_bounds__(256) void k4_apply_mask(
    const float* __restrict__ chmask, float* __restrict__ out)
{
  const int nc = blockIdx.x;
  if (chmask[nc] != 0.f) return;                     // selected channels already final
  const int t = threadIdx.x;
  float4 z = make_float4(0.f, 0.f, 0.f, 0.f);
  float4* dst = (float4*)(out + (size_t)nc * HW);
#pragma unroll
  for (int q = 0; q < 4; ++q) dst[t * 4 + q] = z;
}

// ---------------------------------------------------------------------------
// K5: fold partials into the scalar regularizer.
// ---------------------------------------------------------------------------
__global__ void k5_final(const float* __restrict__ covpart, const float* __restrict__ entW,
                         const float* __restrict__ Ssum, float* __restrict__ out_reg)
{
  if (threadIdx.x == 0 && blockIdx.x == 0) {
    float cov = 0.f, ew = 0.f, ss = 0.f;
    for (int n = 0; n < NBATCH; ++n) { cov += covpart[n]; ew += entW[n]; ss += Ssum[n]; }
    float reg = cov / ((float)NBATCH * HW)                       // LAM_COVERAGE = 1
              + ss  / ((float)NBATCH * NCH * HW)                 // LAM_L1 = 1 (mean |x|)
              - 0.5f * (ew / ((float)NBATCH * NCH));             // LAM_LOCALITY = 0.5
    *out_reg = reg;
  }
}

extern "C" void kernel_launch(void* const* d_in, const int* in_sizes, int n_in,
                              void* d_out, int out_size, void* d_ws, size_t ws_size,
                              hipStream_t stream) {
  const float* x = (const float*)d_in[0];            // (32,256,64,64) f32; tau unused
  float* out = (float*)d_out;                        // [0..33554431]=sparse_x, [33554432]=reg
  float* W = (float*)d_ws;
  float* sparse_ch = W;                              // 8192
  float* Sarr      = W + 8192;                       // 8192
  float* entArr    = W + 16384;                      // 8192
  float* chmask    = W + 24576;                      // 8192
  float* wch       = W + 32768;                      // 8192
  float* entW      = W + 40960;                      // 32
  float* Ssum      = W + 40992;                      // 32
  float* covpart   = W + 41024;                      // 32

  k1_spatial_topk<<<NBATCH * NCH, 256, 0, stream>>>(x, out, sparse_ch, Sarr, entArr);
  k2_channel_topk<<<NBATCH, 256, 0, stream>>>(sparse_ch, Sarr, entArr, chmask, wch, entW, Ssum);
  k3_coverage_wmma<<<NBATCH, 256, 0, stream>>>(x, wch, covpart);
  k4_apply_mask<<<NBATCH * NCH, 256, 0, stream>>>(chmask, out);
  k5_final<<<1, 64, 0, stream>>>(covpart, entW, Ssum, out + (size_t)NBATCH * NCH * HW);
}